// MultiTaskContactPerceiver_15676630630607
// MI455X (gfx1250) — compile-verified
//
#include <hip/hip_runtime.h>
#include <float.h>
#include <math.h>

// Problem constants from the reference
#define BB   4
#define NTOK 2048
#define DD   256
#define NBK  16
#define HEADS 8

typedef float v2f __attribute__((ext_vector_type(2)));
typedef float v8f __attribute__((ext_vector_type(8)));

__device__ __forceinline__ float gelu_f(float x) {
  return 0.5f * x * (1.0f + erff(x * 0.70710678118654752f));
}
__device__ __forceinline__ float sigmoid_f(float x) {
  return 1.0f / (1.0f + __expf(-x));
}

// ---------------------------------------------------------------------------
// WMMA GEMM: C[M,Nc] = A[M,K] @ W[K,Nc] + bias, optional activation.
// fp32 WMMA (V_WMMA_F32_16X16X4_F32). One wave computes a 16x32 tile
// (two accumulators share one A fragment). CONTRACT: K % 4 == 0,
// Nc % 32 == 0, M % 16 == 0, A rows 8-byte aligned (K even).
// grid = (Nc/32, M/16), block = 32.
// ---------------------------------------------------------------------------
__global__ void wmma_gemm_kernel(const float* __restrict__ A,
                                 const float* __restrict__ W,
                                 const float* __restrict__ bias,
                                 float* __restrict__ C,
                                 int K, int Nc, int act) {
  const int tn   = blockIdx.x * 32;
  const int tm   = blockIdx.y * 16;
  const int lane = threadIdx.x;
  const int li   = lane & 15;
  const int half = lane >> 4;
  const float* Arow = A + (size_t)(tm + li) * K + 2 * half;
  const float* Wc0  = W + tn + li;
  const float* Wc1  = W + tn + 16 + li;
  v8f acc0 = {0.f, 0.f, 0.f, 0.f, 0.f, 0.f, 0.f, 0.f};
  v8f acc1 = {0.f, 0.f, 0.f, 0.f, 0.f, 0.f, 0.f, 0.f};
#pragma unroll 4
  for (int k0 = 0; k0 < K; k0 += 4) {
    const int ka = k0 + 2 * half;
    const v2f a = *(const v2f*)(Arow + k0);
    v2f b0, b1;
    b0.x = Wc0[(size_t)ka * Nc];
    b0.y = Wc0[(size_t)(ka + 1) * Nc];
    b1.x = Wc1[(size_t)ka * Nc];
    b1.y = Wc1[(size_t)(ka + 1) * Nc];
    acc0 = __builtin_amdgcn_wmma_f32_16x16x4_f32(false, a, false, b0,
                                                 (short)0, acc0, false, false);
    acc1 = __builtin_amdgcn_wmma_f32_16x16x4_f32(false, a, false, b1,
                                                 (short)0, acc1, false, false);
  }
  const int col0 = tn + li;
  const float bv0 = bias ? bias[col0] : 0.0f;
  const float bv1 = bias ? bias[col0 + 16] : 0.0f;
#pragma unroll
  for (int r = 0; r < 8; ++r) {
    float* Crow = C + (size_t)(tm + r + 8 * half) * Nc;
    float v0 = acc0[r] + bv0;
    float v1 = acc1[r] + bv1;
    if (act == 1) { v0 = gelu_f(v0); v1 = gelu_f(v1); }
    else if (act == 2) { v0 = sigmoid_f(v0); v1 = sigmoid_f(v1); }
    Crow[col0] = v0;
    Crow[col0 + 16] = v1;
  }
}

// ---------------------------------------------------------------------------
// Scalar GEMM (tiny M or tiny K paths: text branch, CA query path, spatial).
// ---------------------------------------------------------------------------
__global__ void rowgemm_kernel(const float* __restrict__ A,
                               const float* __restrict__ W,
                               const float* __restrict__ bias,
                               float* __restrict__ C,
                               int M, int K, int Nc, int act) {
  const int idx = blockIdx.x * 256 + threadIdx.x;
  if (idx >= M * Nc) return;
  const int m = idx / Nc;
  const int j = idx - m * Nc;
  float s = bias ? bias[j] : 0.0f;
  const float* a = A + (size_t)m * K;
  for (int k = 0; k < K; ++k) s += a[k] * W[(size_t)k * Nc + j];
  if (act == 1) s = gelu_f(s);
  else if (act == 2) s = sigmoid_f(s);
  C[idx] = s;
}

// ---------------------------------------------------------------------------
// LayerNorm over D=256 (one block per row), optional trailing GELU.
// ---------------------------------------------------------------------------
__global__ void ln_kernel(const float* __restrict__ X,
                          const float* __restrict__ g,
                          const float* __restrict__ b,
                          float* __restrict__ Y, int act) {
  const int row = blockIdx.x;
  const int d   = threadIdx.x;
  __shared__ float sh[256];
  const float xv = X[(size_t)row * 256 + d];
  sh[d] = xv; __syncthreads();
  for (int s = 128; s > 0; s >>= 1) { if (d < s) sh[d] += sh[d + s]; __syncthreads(); }
  const float mean = sh[0] * (1.0f / 256.0f);
  __syncthreads();
  const float c = xv - mean;
  sh[d] = c * c; __syncthreads();
  for (int s = 128; s > 0; s >>= 1) { if (d < s) sh[d] += sh[d + s]; __syncthreads(); }
  const float var = sh[0] * (1.0f / 256.0f);
  float y = c / sqrtf(var + 1e-5f) * g[d] + b[d];
  if (act == 1) y = gelu_f(y);
  Y[(size_t)row * 256 + d] = y;
}

// ---------------------------------------------------------------------------
// Per-batch xyz stats: mean xyz, ymin, ymax. grid=B, block=256.
// stats layout per b: [0..2]=mean, [3]=ymin, [4]=ymax (stride 8).
// ---------------------------------------------------------------------------
__global__ void stats_kernel(const float* __restrict__ xyz, float* __restrict__ stats) {
  const int b = blockIdx.x, t = threadIdx.x;
  __shared__ float sh[256];
  float ax = 0, ay = 0, az = 0, mn = FLT_MAX, mx = -FLT_MAX;
  for (int n = t; n < NTOK; n += 256) {
    const float* p = xyz + (size_t)(b * NTOK + n) * 3;
    ax += p[0]; const float y = p[1]; ay += y; az += p[2];
    mn = fminf(mn, y); mx = fmaxf(mx, y);
  }
  float vals[5] = {ax, ay, az, mn, mx};
#pragma unroll
  for (int k = 0; k < 5; ++k) {
    sh[t] = vals[k]; __syncthreads();
    for (int s = 128; s > 0; s >>= 1) {
      if (t < s) {
        if (k < 3) sh[t] += sh[t + s];
        else if (k == 3) sh[t] = fminf(sh[t], sh[t + s]);
        else sh[t] = fmaxf(sh[t], sh[t + s]);
      }
      __syncthreads();
    }
    if (t == 0) stats[b * 8 + k] = (k < 3) ? sh[0] * (1.0f / NTOK) : sh[0];
    __syncthreads();
  }
}

__global__ void sq_kernel(const float* __restrict__ xyz, float* __restrict__ sq) {
  const int i = blockIdx.x * 256 + threadIdx.x;
  if (i < BB * NTOK) {
    const float* p = xyz + (size_t)i * 3;
    sq[i] = p[0] * p[0] + p[1] * p[1] + p[2] * p[2];
  }
}

__global__ void concat_kernel(const float* __restrict__ x, const float* __restrict__ pf,
                              const float* __restrict__ xyz, float* __restrict__ A) {
  const int i = blockIdx.x * 256 + threadIdx.x;
  const int total = BB * NTOK * 516;
  if (i >= total) return;
  const int row = i / 516;
  const int k = i - row * 516;
  float v;
  if (k == 0)        v = x[row];
  else if (k < 513)  v = pf[(size_t)row * 512 + (k - 1)];
  else               v = xyz[(size_t)row * 3 + (k - 513)];
  A[i] = v;
}

// ---------------------------------------------------------------------------
// kNN(16) + region mean + spatial features. One block (256 thr) per point.
// ---------------------------------------------------------------------------
__global__ void knn_kernel(const float* __restrict__ xyz, const float* __restrict__ sq,
                           const float* __restrict__ stats, const float* __restrict__ local,
                           float* __restrict__ region_in, float* __restrict__ sp6) {
  const int row = blockIdx.x;
  const int b = row / NTOK;
  const int n = row - b * NTOK;
  const int t = threadIdx.x;
  __shared__ float d2s[NTOK];
  __shared__ float red[256];
  __shared__ int   redi[256];
  __shared__ int   sel[NBK];
  const float* pb = xyz + (size_t)(b * NTOK) * 3;
  const float px = pb[n * 3 + 0], py = pb[n * 3 + 1], pz = pb[n * 3 + 2];
  const float sqn = sq[row];
  const float* sqb = sq + b * NTOK;
  float rsum = 0.0f, rmin = FLT_MAX, rmax = -FLT_MAX;
  for (int m = t; m < NTOK; m += 256) {
    const float qx = pb[m * 3], qy = pb[m * 3 + 1], qz = pb[m * 3 + 2];
    float d2 = sqn + sqb[m] - 2.0f * (px * qx + py * qy + pz * qz);
    d2 = fmaxf(d2, 0.0f);
    d2s[m] = d2;
    const float rd = sqrtf(fmaxf(d2, 1e-12f));
    rsum += rd; rmin = fminf(rmin, rd); rmax = fmaxf(rmax, rd);
  }
  red[t] = rsum; __syncthreads();
  for (int s = 128; s > 0; s >>= 1) { if (t < s) red[t] += red[t + s]; __syncthreads(); }
  const float rmean = red[0] * (1.0f / NTOK); __syncthreads();
  red[t] = rmin; __syncthreads();
  for (int s = 128; s > 0; s >>= 1) { if (t < s) red[t] = fminf(red[t], red[t + s]); __syncthreads(); }
  const float rminv = red[0]; __syncthreads();
  red[t] = rmax; __syncthreads();
  for (int s = 128; s > 0; s >>= 1) { if (t < s) red[t] = fmaxf(red[t], red[t + s]); __syncthreads(); }
  const float rmaxv = red[0]; __syncthreads();
  // top-16 smallest d2 (ties -> lowest index, matching jax top_k on -d2)
  for (int it = 0; it < NBK; ++it) {
    float bv = FLT_MAX; int bi = 0x7fffffff;
    for (int m = t; m < NTOK; m += 256) {
      const float v = d2s[m];
      if (v < bv) { bv = v; bi = m; }
    }
    red[t] = bv; redi[t] = bi; __syncthreads();
    for (int s = 128; s > 0; s >>= 1) {
      if (t < s) {
        if (red[t + s] < red[t] || (red[t + s] == red[t] && redi[t + s] < redi[t])) {
          red[t] = red[t + s]; redi[t] = redi[t + s];
        }
      }
      __syncthreads();
    }
    if (t == 0) { sel[it] = redi[0]; d2s[redi[0]] = FLT_MAX; }
    __syncthreads();
  }
  float acc = 0.0f;
#pragma unroll
  for (int j = 0; j < NBK; ++j)
    acc += local[((size_t)(b * NTOK + sel[j])) * DD + t];
  region_in[(size_t)row * DD + t] = acc * (1.0f / NBK);
  if (t < 3)      sp6[(size_t)row * 6 + t] = ((t == 0) ? px : (t == 1) ? py : pz) - stats[b * 8 + t];
  else if (t == 3) sp6[(size_t)row * 6 + 3] = rmean;
  else if (t == 4) sp6[(size_t)row * 6 + 4] = rmaxv;
  else if (t == 5) sp6[(size_t)row * 6 + 5] = rminv;
}

__global__ void lmean_kernel(const float* __restrict__ L, float* __restrict__ out) {
  const int b = blockIdx.x, d = threadIdx.x;
  float s = 0.0f;
  const float* base = L + (size_t)b * NTOK * DD + d;
  for (int n = 0; n < NTOK; ++n) s += base[(size_t)n * DD];
  out[b * DD + d] = s * (1.0f / NTOK);
}

__global__ void add3_kernel(const float* __restrict__ A, const float* __restrict__ Bm,
                            const float* __restrict__ G, float* __restrict__ C, int total) {
  const int i = blockIdx.x * 256 + threadIdx.x;
  if (i < total) {
    const int row = i >> 8;
    const int b = row >> 11;  // N=2048
    C[i] = A[i] + Bm[i] + G[(b << 8) + (i & 255)];
  }
}

__global__ void add2_kernel(const float* __restrict__ A, const float* __restrict__ Bm,
                            float* __restrict__ C, int total) {
  const int i = blockIdx.x * 256 + threadIdx.x;
  if (i < total) C[i] = A[i] + Bm[i];
}

__global__ void gated_kernel(float* __restrict__ q, const float* __restrict__ gate,
                             const float* __restrict__ outv, int total) {
  const int i = blockIdx.x * 256 + threadIdx.x;
  if (i < total) q[i] = q[i] + gate[i] * outv[i];
}

// ---------------------------------------------------------------------------
// Flash attention, 8 heads of hd=32, fp32 WMMA. One wave per 16-query tile.
// grid = (N/16, H, B), block = 32.
// ---------------------------------------------------------------------------
__global__ void flash_attn_kernel(const float* __restrict__ Q, const float* __restrict__ Kt,
                                  const float* __restrict__ V, float* __restrict__ O) {
  const int qt = blockIdx.x * 16;
  const int h  = blockIdx.y;
  const int b  = blockIdx.z;
  const int lane = threadIdx.x;
  const int li = lane & 15, half = lane >> 4;
  const float scale = 0.17677669529663687f;  // 1/sqrt(32)
  __shared__ float Pld[16][18];              // even stride -> b64-aligned reads

  const float* Qrow = Q + ((size_t)(b * NTOK + qt + li)) * DD + h * 32 + 2 * half;
  v2f qa[8];
#pragma unroll
  for (int s = 0; s < 8; ++s) qa[s] = *(const v2f*)(Qrow + 4 * s);
  float mstate[8], lstate[8];
#pragma unroll
  for (int r = 0; r < 8; ++r) { mstate[r] = -FLT_MAX; lstate[r] = 0.0f; }
  v8f acc0 = {0.f,0.f,0.f,0.f,0.f,0.f,0.f,0.f};
  v8f acc1 = {0.f,0.f,0.f,0.f,0.f,0.f,0.f,0.f};

  for (int kt = 0; kt < NTOK; kt += 16) {
    v8f S = {0.f,0.f,0.f,0.f,0.f,0.f,0.f,0.f};
    const float* Krow = Kt + ((size_t)(b * NTOK + kt + li)) * DD + h * 32 + 2 * half;
#pragma unroll
    for (int s = 0; s < 8; ++s) {
      const v2f bb = *(const v2f*)(Krow + 4 * s);
      S = __builtin_amdgcn_wmma_f32_16x16x4_f32(false, qa[s], false, bb,
                                                (short)0, S, false, false);
    }
#pragma unroll
    for (int r = 0; r < 8; ++r) {
      const float v = S[r] * scale;
      float mrow = v;
      mrow = fmaxf(mrow, __shfl_xor(mrow, 1, 32));
      mrow = fmaxf(mrow, __shfl_xor(mrow, 2, 32));
      mrow = fmaxf(mrow, __shfl_xor(mrow, 4, 32));
      mrow = fmaxf(mrow, __shfl_xor(mrow, 8, 32));
      const float mnew = fmaxf(mstate[r], mrow);
      const float p = __expf(v - mnew);
      float srow = p;
      srow += __shfl_xor(srow, 1, 32);
      srow += __shfl_xor(srow, 2, 32);
      srow += __shfl_xor(srow, 4, 32);
      srow += __shfl_xor(srow, 8, 32);
      const float alpha = __expf(mstate[r] - mnew);
      lstate[r] = lstate[r] * alpha + srow;
      mstate[r] = mnew;
      acc0[r] *= alpha;
      acc1[r] *= alpha;
      Pld[r + 8 * half][li] = p;
    }
    __syncthreads();
#pragma unroll
    for (int s = 0; s < 4; ++s) {
      const v2f pa = *(const v2f*)(&Pld[li][4 * s + 2 * half]);
      const float* Vr = V + ((size_t)(b * NTOK + kt + 4 * s + 2 * half)) * DD + h * 32 + li;
      v2f vb0; vb0.x = Vr[0];  vb0.y = Vr[DD];
      acc0 = __builtin_amdgcn_wmma_f32_16x16x4_f32(false, pa, false, vb0,
                                                   (short)0, acc0, false, false);
      v2f vb1; vb1.x = Vr[16]; vb1.y = Vr[DD + 16];
      acc1 = __builtin_amdgcn_wmma_f32_16x16x4_f32(false, pa, false, vb1,
                                                   (short)0, acc1, false, false);
    }
    __syncthreads();
  }
#pragma unroll
  for (int r = 0; r < 8; ++r) {
    const float invl = 1.0f / lstate[r];
    float* Orow = O + ((size_t)(b * NTOK + qt + r + 8 * half)) * DD + h * 32;
    Orow[li]      = acc0[r] * invl;
    Orow[16 + li] = acc1[r] * invl;
  }
}

// ---------------------------------------------------------------------------
// Cross-attention with query length 1. grid = (H, B), block = 256.
// ---------------------------------------------------------------------------
__global__ void ca_attn_kernel(const float* __restrict__ qh, const float* __restrict__ kh,
                               const float* __restrict__ vh, float* __restrict__ o) {
  const int h = blockIdx.x, b = blockIdx.y, t = threadIdx.x;
  __shared__ float lg[NTOK];
  __shared__ float red[256];
  __shared__ float qs[32];
  if (t < 32) qs[t] = qh[b * 256 + h * 32 + t];
  __syncthreads();
  float mx = -FLT_MAX;
  for (int m = t; m < NTOK; m += 256) {
    const float* kr = kh + ((size_t)(b * NTOK + m)) * DD + h * 32;
    float dv = 0.0f;
#pragma unroll
    for (int d = 0; d < 32; ++d) dv += qs[d] * kr[d];
    dv *= 0.17677669529663687f;
    lg[m] = dv; mx = fmaxf(mx, dv);
  }
  red[t] = mx; __syncthreads();
  for (int s = 128; s > 0; s >>= 1) { if (t < s) red[t] = fmaxf(red[t], red[t + s]); __syncthreads(); }
  mx = red[0]; __syncthreads();
  float sm = 0.0f;
  for (int m = t; m < NTOK; m += 256) {
    const float p = __expf(lg[m] - mx);
    lg[m] = p; sm += p;
  }
  red[t] = sm; __syncthreads();
  for (int s = 128; s > 0; s >>= 1) { if (t < s) red[t] += red[t + s]; __syncthreads(); }
  const float inv = 1.0f / red[0]; __syncthreads();
  const int d = t & 31, grp = t >> 5;
  float acc = 0.0f;
  for (int m = grp; m < NTOK; m += 8)
    acc += lg[m] * vh[((size_t)(b * NTOK + m)) * DD + h * 32 + d];
  red[t] = acc; __syncthreads();
  if (t < 32) {
    float s = 0.0f;
#pragma unroll
    for (int g2 = 0; g2 < 8; ++g2) s += red[g2 * 32 + t];
    o[b * 256 + h * 32 + t] = s * inv;
  }
}

// ---------------------------------------------------------------------------
// Physical constraints: sp2, gw, cp, broadcast of fused. grid=B*N, block=256.
// ---------------------------------------------------------------------------
__global__ void phys_kernel(const float* __restrict__ fused, const float* __restrict__ hA,
                            const float* __restrict__ s1w, const float* __restrict__ s1b,
                            const float* __restrict__ s2w, const float* __restrict__ s2b,
                            const float* __restrict__ cp, const float* __restrict__ xyz,
                            const float* __restrict__ stats, float* __restrict__ C) {
  const int row = blockIdx.x;
  const int d = threadIdx.x;
  const int b = row / NTOK;
  __shared__ float red[256];
  __shared__ float fshare;
  const float* p = xyz + (size_t)row * 3;
  float t = hA[b * 256 + d] + p[0] * s1w[256 * 256 + d] + p[1] * s1w[257 * 256 + d] +
            p[2] * s1w[258 * 256 + d] + s1b[d];
  t = gelu_f(t);
  red[d] = t * s2w[d]; __syncthreads();
  for (int s = 128; s > 0; s >>= 1) { if (d < s) red[d] += red[d + s]; __syncthreads(); }
  if (d == 0) {
    const float sp2 = sigmoid_f(red[0] + s2b[0]);
    const float ymin = stats[b * 8 + 3], ymax = stats[b * 8 + 4];
    const float gw = 1.0f - (p[1] - ymin) / (ymax - ymin + 1e-8f);
    fshare = cp[b] * gw * sp2;
  }
  __syncthreads();
  C[(size_t)row * 256 + d] = fused[b * 256 + d] * fshare;
}

__global__ void head_kernel(const float* __restrict__ F, const float* __restrict__ w,
                            const float* __restrict__ bsc, float* __restrict__ out) {
  const int row = blockIdx.x, d = threadIdx.x;
  __shared__ float red[256];
  red[d] = F[(size_t)row * 256 + d] * w[d]; __syncthreads();
  for (int s = 128; s > 0; s >>= 1) { if (d < s) red[d] += red[d + s]; __syncthreads(); }
  if (d == 0) out[row] = sigmoid_f(red[0] + bsc[0]);
}

// ---------------------------------------------------------------------------
// Host launcher
// ---------------------------------------------------------------------------
static inline unsigned cdiv(unsigned a, unsigned b) { return (a + b - 1) / b; }

extern "C" void kernel_launch(void* const* d_in, const int* in_sizes, int n_in,
                              void* d_out, int out_size, void* d_ws, size_t ws_size,
                              hipStream_t stream) {
  const float* x    = (const float*)d_in[0];
  const float* pf   = (const float*)d_in[1];
  const float* lang = (const float*)d_in[2];
  const float* xyz  = (const float*)d_in[4];

  // ---- param leaves in jax pytree (sorted-key) order --------------------
  enum { NLEAF = 124 };
  int lsz[NLEAF];
  {
    int t = 0;
    const int ca[24] = {256,65536,256,65536,256,65536,256,65536,   // attn k/o/q/v (b,w)
                        256,65536,256,65536,                        // g1, g2
                        256,256, 256,256, 256,256, 256,256,        // gn, ln2, lnkv, lnq
                        512,131072, 256,131072};                    // m1, m2
    for (int l = 0; l < 3; ++l) for (int j = 0; j < 24; ++j) lsz[t++] = ca[j];
    lsz[t++] = 1; lsz[t++] = 256;                                   // head
    const int ou[6] = {256,65536,256,65536,256,256};               // out l1,l2,n
    for (int j = 0; j < 6; ++j) lsz[t++] = ou[j];
    const int pe[28] = {256,65536,256,65536,256,65536,256,65536,   // pe.attn
                        256,65536,256,256,                          // glob l,n
                        256,132096,256,256,                         // local l(516x256),n
                        256,65536,256,65536,256,256,256,256,        // region l1,l2,n1,n2
                        256,1536,256,256};                          // spatial l(6x256),n
    for (int j = 0; j < 28; ++j) lsz[t++] = pe[j];
    const int ph[8] = {256,65536,1,256,256,66304,1,256};           // c1,c2,s1(259x256),s2
    for (int j = 0; j < 8; ++j) lsz[t++] = ph[j];
    const int tx[8] = {256,131072,256,65536,256,256,256,256};      // text l1,l2,n1,n2
    for (int j = 0; j < 8; ++j) lsz[t++] = tx[j];
  }
  const float* P[NLEAF];
  if (n_in >= 5 + NLEAF) {
    for (int i = 0; i < NLEAF; ++i) P[i] = (const float*)d_in[5 + i];
  } else {
    const float* base = (const float*)d_in[5];
    size_t off = 0;
    for (int i = 0; i < NLEAF; ++i) { P[i] = base + off; off += lsz[i]; }
  }
  // named indices
  const int HEADB=72, HEADW=73;
  const int OL1B=74, OL1W=75, OL2B=76, OL2W=77, ONB=78, ONG=79;
  const int AKB=80, AKW=81, AOB=82, AOW=83, AQB=84, AQW=85, AVB=86, AVW=87;
  const int GLB=88, GLW=89, GNB=90, GNG=91;
  const int LLB=92, LLW=93, LNB=94, LNG=95;
  const int R1B=96, R1W=97, R2B=98, R2W=99, RN1B=100, RN1G=101, RN2B=102, RN2G=103;
  const int SLB=104, SLW=105, SNB=106, SNG=107;
  const int C1B=108, C1W=109, C2B=110, C2W=111, S1B=112, S1W=113, S2B=114, S2W=115;
  const int TL1B=116, TL1W=117, TL2B=118, TL2W=119, TN1B=120, TN1G=121, TN2B=122, TN2G=123;

  // ---- workspace layout -------------------------------------------------
  const size_t MD = (size_t)BB * NTOK * DD;  // 2,097,152
  float* w0 = (float*)d_ws;
  float* ACONC = w0;                        // 8192*516 (also reused: QH, KH)
  float* L   = ACONC + (size_t)BB * NTOK * 516;
  float* Rg0 = L   + MD;
  float* Rg1 = Rg0 + MD;
  float* Vb  = Rg1 + MD;
  float* Pb  = Vb  + MD;   // pemb
  float* KNb = Pb  + MD;
  float* KH2 = KNb + MD;
  float* VH2 = KH2 + MD;
  float* Tb  = VH2 + MD;   // constrained temp
  float* SQb = Tb  + MD;                    // 8192
  float* SP6 = SQb + (size_t)BB * NTOK;     // 8192*6
  float* SM  = SP6 + (size_t)BB * NTOK * 6; // 32 slots * 2048 floats
#define SLOT(i) (SM + (size_t)(i) * 2048)
  float* STATS = SLOT(20);
  float* QH = ACONC;
  float* KH = ACONC + MD;

  const dim3 b256(256), b32(32);
  const dim3 gGemm(DD / 32, (BB * NTOK) / 16);  // (8, 512)
  const int MDtot = (int)MD;

  // ---- point / stats ----------------------------------------------------
  stats_kernel<<<dim3(BB), b256, 0, stream>>>(xyz, STATS);
  sq_kernel<<<dim3(cdiv(BB * NTOK, 256)), b256, 0, stream>>>(xyz, SQb);
  concat_kernel<<<dim3(cdiv(BB * NTOK * 516, 256)), b256, 0, stream>>>(x, pf, xyz, ACONC);

  // local = gelu(LN(concat @ Wl + b))
  wmma_gemm_kernel<<<gGemm, b32, 0, stream>>>(ACONC, P[LLW], P[LLB], L, 516, DD, 0);
  ln_kernel<<<dim3(BB * NTOK), b256, 0, stream>>>(L, P[LNG], P[LNB], L, 1);

  // kNN + region_in + spatial features
  knn_kernel<<<dim3(BB * NTOK), b256, 0, stream>>>(xyz, SQb, STATS, L, Rg0, SP6);

  // region MLP
  wmma_gemm_kernel<<<gGemm, b32, 0, stream>>>(Rg0, P[R1W], P[R1B], Rg1, DD, DD, 0);
  ln_kernel<<<dim3(BB * NTOK), b256, 0, stream>>>(Rg1, P[RN1G], P[RN1B], Rg1, 1);
  wmma_gemm_kernel<<<gGemm, b32, 0, stream>>>(Rg1, P[R2W], P[R2B], Rg0, DD, DD, 0);
  ln_kernel<<<dim3(BB * NTOK), b256, 0, stream>>>(Rg0, P[RN2G], P[RN2B], Rg0, 0);

  // glob
  lmean_kernel<<<dim3(BB), b256, 0, stream>>>(L, SLOT(16));
  rowgemm_kernel<<<dim3(cdiv(BB * DD, 256)), b256, 0, stream>>>(SLOT(16), P[GLW], P[GLB], SLOT(17), BB, DD, DD, 0);
  ln_kernel<<<dim3(BB), b256, 0, stream>>>(SLOT(17), P[GNG], P[GNB], SLOT(17), 0);

  // combined = local + region + gf
  add3_kernel<<<dim3(cdiv(MDtot, 256)), b256, 0, stream>>>(L, Rg0, SLOT(17), Rg1, MDtot);

  // self-attention: projections + flash + output projection
  wmma_gemm_kernel<<<gGemm, b32, 0, stream>>>(Rg1, P[AQW], P[AQB], QH, DD, DD, 0);
  wmma_gemm_kernel<<<gGemm, b32, 0, stream>>>(Rg1, P[AKW], P[AKB], KH, DD, DD, 0);
  wmma_gemm_kernel<<<gGemm, b32, 0, stream>>>(Rg1, P[AVW], P[AVB], Vb, DD, DD, 0);
  flash_attn_kernel<<<dim3(NTOK / 16, HEADS, BB), b32, 0, stream>>>(QH, KH, Vb, Rg0);
  wmma_gemm_kernel<<<gGemm, b32, 0, stream>>>(Rg0, P[AOW], P[AOB], Pb, DD, DD, 0);

  // spatial branch (K=6 -> scalar path), pemb = attended + spatial
  rowgemm_kernel<<<dim3(cdiv(BB * NTOK * DD, 256)), b256, 0, stream>>>(SP6, P[SLW], P[SLB], KNb, BB * NTOK, 6, DD, 0);
  ln_kernel<<<dim3(BB * NTOK), b256, 0, stream>>>(KNb, P[SNG], P[SNB], KNb, 1);
  add2_kernel<<<dim3(cdiv(MDtot, 256)), b256, 0, stream>>>(Pb, KNb, Pb, MDtot);

  // text branch -> q (SLOT 0)
  rowgemm_kernel<<<dim3(cdiv(BB * DD, 256)), b256, 0, stream>>>(lang, P[TL1W], P[TL1B], SLOT(12), BB, 512, DD, 0);
  ln_kernel<<<dim3(BB), b256, 0, stream>>>(SLOT(12), P[TN1G], P[TN1B], SLOT(12), 1);
  rowgemm_kernel<<<dim3(cdiv(BB * DD, 256)), b256, 0, stream>>>(SLOT(12), P[TL2W], P[TL2B], SLOT(0), BB, DD, DD, 0);
  ln_kernel<<<dim3(BB), b256, 0, stream>>>(SLOT(0), P[TN2G], P[TN2B], SLOT(0), 0);

  // cross-attention layers
  for (int l = 0; l < 3; ++l) {
    const int c = l * 24;
    ln_kernel<<<dim3(BB), b256, 0, stream>>>(SLOT(0), P[c + 19], P[c + 18], SLOT(1), 0);        // qn
    ln_kernel<<<dim3(BB * NTOK), b256, 0, stream>>>(Pb, P[c + 17], P[c + 16], KNb, 0);          // kn
    rowgemm_kernel<<<dim3(cdiv(BB * DD, 256)), b256, 0, stream>>>(SLOT(1), P[c + 5], P[c + 4], SLOT(2), BB, DD, DD, 0);
    wmma_gemm_kernel<<<gGemm, b32, 0, stream>>>(KNb, P[c + 1], P[c + 0], KH2, DD, DD, 0);
    wmma_gemm_kernel<<<gGemm, b32, 0, stream>>>(KNb, P[c + 7], P[c + 6], VH2, DD, DD, 0);
    ca_attn_kernel<<<dim3(HEADS, BB), b256, 0, stream>>>(SLOT(2), KH2, VH2, SLOT(3));
    rowgemm_kernel<<<dim3(cdiv(BB * DD, 256)), b256, 0, stream>>>(SLOT(3), P[c + 3], P[c + 2], SLOT(4), BB, DD, DD, 0);
    add2_kernel<<<dim3(cdiv(BB * DD, 256)), b256, 0, stream>>>(SLOT(0), SLOT(4), SLOT(5), BB * DD);   // h
    ln_kernel<<<dim3(BB), b256, 0, stream>>>(SLOT(5), P[c + 15], P[c + 14], SLOT(6), 0);
    rowgemm_kernel<<<dim3(cdiv(BB * 512, 256)), b256, 0, stream>>>(SLOT(6), P[c + 21], P[c + 20], SLOT(7), BB, DD, 512, 1);
    rowgemm_kernel<<<dim3(cdiv(BB * DD, 256)), b256, 0, stream>>>(SLOT(7), P[c + 23], P[c + 22], SLOT(8), BB, 512, DD, 0);
    add2_kernel<<<dim3(cdiv(BB * DD, 256)), b256, 0, stream>>>(SLOT(5), SLOT(8), SLOT(9), BB * DD);   // out
    rowgemm_kernel<<<dim3(cdiv(BB * DD, 256)), b256, 0, stream>>>(SLOT(0), P[c + 9], P[c + 8], SLOT(10), BB, DD, DD, 0);
    ln_kernel<<<dim3(BB), b256, 0, stream>>>(SLOT(10), P[c + 13], P[c + 12], SLOT(10), 1);
    rowgemm_kernel<<<dim3(cdiv(BB * DD, 256)), b256, 0, stream>>>(SLOT(10), P[c + 11], P[c + 10], SLOT(11), BB, DD, DD, 2); // gate
    gated_kernel<<<dim3(cdiv(BB * DD, 256)), b256, 0, stream>>>(SLOT(0), SLOT(11), SLOT(9), BB * DD);
  }

  // physics-constrained broadcast
  rowgemm_kernel<<<dim3(cdiv(BB * DD, 256)), b256, 0, stream>>>(SLOT(0), P[C1W], P[C1B], SLOT(13), BB, DD, DD, 1);
  rowgemm_kernel<<<dim3(1), b256, 0, stream>>>(SLOT(13), P[C2W], P[C2B], SLOT(14), BB, DD, 1, 2);  // cp
  rowgemm_kernel<<<dim3(cdiv(BB * DD, 256)), b256, 0, stream>>>(SLOT(0), P[S1W], nullptr, SLOT(15), BB, DD, DD, 0); // hA
  phys_kernel<<<dim3(BB * NTOK), b256, 0, stream>>>(SLOT(0), SLOT(15), P[S1W], P[S1B],
                                                    P[S2W], P[S2B], SLOT(14), xyz, STATS, Tb);

  // output MLP + head
  wmma_gemm_kernel<<<gGemm, b32, 0, stream>>>(Tb, P[OL1W], P[OL1B], KNb, DD, DD, 0);
  ln_kernel<<<dim3(BB * NTOK), b256, 0, stream>>>(KNb, P[ONG], P[ONB], KNb, 1);
  wmma_gemm_kernel<<<gGemm, b32, 0, stream>>>(KNb, P[OL2W], P[OL2B], Rg0, DD, DD, 0);
  head_kernel<<<dim3(BB * NTOK), b256, 0, stream>>>(Rg0, P[HEADW], P[HEADB], (float*)d_out);
  (void)in_sizes; (void)out_size; (void)ws_size;
}